// SpatialPosEncoder_35107062677894
// MI455X (gfx1250) — compile-verified
//
#include <hip/hip_runtime.h>
#include <math.h>

#define NRES 512
#define CZ 128
#define WPACK_FLOATS (512 * 128)                 /* 65536 floats: 4 blocks of 128 used W-columns, bias folded */
#define CA_FLOATS (NRES * 3)                     /* 1536 */
#define STAGE_FLOATS (WPACK_FLOATS + CA_FLOATS)  /* 67072 floats = 268288 B, fits 320KB LDS */
#define STAGE_VEC4 (STAGE_FLOATS / 4)            /* 16768 b128 transfers */

typedef float v4f __attribute__((ext_vector_type(4)));

// ---------------------------------------------------------------------------
// Pack W (128 x 516, row-major) into wpack (512 rows x 128 ch):
//   row = t*128 + j  ->  feature f = t*129 + 1 + j  (index 0 of each one-hot
//   block is unreachable: argmax+1 >= 1). Bias folded into block t==0.
// ---------------------------------------------------------------------------
__global__ void pack_w_kernel(const float* __restrict__ W,
                              const float* __restrict__ b,
                              float* __restrict__ wpack) {
    int idx = blockIdx.x * blockDim.x + threadIdx.x;
    if (idx >= 512 * 128) return;
    int row = idx >> 7;          // 0..511
    int c   = idx & 127;         // channel
    int t   = row >> 7;          // which one-hot block (0..3)
    int j   = row & 127;         // bin-1
    int f   = t * 129 + 1 + j;   // feature column in W
    float v = W[c * 516 + f];
    if (t == 0) v += b[c];
    wpack[(row << 7) + c] = v;
}

// ---------------------------------------------------------------------------
// Per-residue frame: x = norm(N-CA), z = norm(cross(x, C-CA)), y = cross(z,x)
// Also extracts ca into a dense array (contiguous after wpack in ws so the
// main kernel stages both with a single async loop).
// ---------------------------------------------------------------------------
__global__ void frames_kernel(const float* __restrict__ pos,
                              float* __restrict__ frames,
                              float* __restrict__ ca_out) {
    int i = blockIdx.x * blockDim.x + threadIdx.x;
    if (i >= NRES) return;
    const float* n  = pos + (size_t)(i * 37 + 0) * 3;
    const float* ca = pos + (size_t)(i * 37 + 1) * 3;
    const float* c  = pos + (size_t)(i * 37 + 2) * 3;
    float cax = ca[0], cay = ca[1], caz = ca[2];
    float x0 = n[0] - cax, x1 = n[1] - cay, x2 = n[2] - caz;
    float nx = fmaxf(sqrtf(x0 * x0 + x1 * x1 + x2 * x2), 1e-12f);
    x0 /= nx; x1 /= nx; x2 /= nx;
    float u0 = c[0] - cax, u1 = c[1] - cay, u2 = c[2] - caz;
    float z0 = x1 * u2 - x2 * u1;
    float z1 = x2 * u0 - x0 * u2;
    float z2 = x0 * u1 - x1 * u0;
    float nz = fmaxf(sqrtf(z0 * z0 + z1 * z1 + z2 * z2), 1e-12f);
    z0 /= nz; z1 /= nz; z2 /= nz;
    float y0 = z1 * x2 - z2 * x1;
    float y1 = z2 * x0 - z0 * x2;
    float y2 = z0 * x1 - z1 * x0;
    float* fr = frames + i * 9;
    fr[0] = x0; fr[1] = x1; fr[2] = x2;
    fr[3] = y0; fr[4] = y1; fr[5] = y2;
    fr[6] = z0; fr[7] = z1; fr[8] = z2;
    float* co = ca_out + i * 3;
    co[0] = cax; co[1] = cay; co[2] = caz;
}

// linspace(-50,50,127) bins with [-1e8 / 1e8] sentinels, strict inequalities:
//   v <= -50 (incl. all-false fallthrough) -> bin 0 ; v > 50 -> bin 127
__device__ __forceinline__ int dir_bin(float v) {
    int t = (int)floorf((v + 50.0f) * (126.0f / 100.0f)) + 1;
    t = t > 127 ? 127 : t;
    return (v <= -50.0f) ? 0 : t;
}

// ---------------------------------------------------------------------------
// Main kernel: 256 blocks x 1024 threads (32 wave32s). Stage wpack+ca into
// LDS once via CDNA5 async global->LDS DMA (ASYNCcnt), then each wave
// produces 512B output rows: 4 x ds_load_b128 gathers + 1 x non-temporal
// global_store_b128 per pair (output is write-once streaming -> NT hint
// keeps the 128MiB result from rinsing through L2).
// Each block serves i = blockIdx.x and blockIdx.x+256 (amortizes staging).
// ---------------------------------------------------------------------------
__global__ __launch_bounds__(1024) void spe_main_kernel(
        const float* __restrict__ ws_f,      // [wpack | ca] contiguous
        const float* __restrict__ frames_g,  // 512 x 9
        float* __restrict__ out) {           // 512 x 512 x 128
    extern __shared__ float smem[];          // STAGE_FLOATS floats
    const int tid = threadIdx.x;

    // ---- async stage: 268288 B global -> LDS (ASYNCcnt-tracked DMA) -------
    uint64_t gbase = (uint64_t)(uintptr_t)ws_f;
    uint32_t lbase = (uint32_t)(uintptr_t)smem;  // low 32b of flat = LDS addr
    for (int idx = tid; idx < STAGE_VEC4; idx += 1024) {
        uint32_t boff  = (uint32_t)idx * 16u;
        uint32_t laddr = lbase + boff;
        uint64_t gaddr = gbase + boff;
        asm volatile("global_load_async_to_lds_b128 %0, %1, off"
                     :: "v"(laddr), "v"(gaddr) : "memory");
    }

    const int wave = tid >> 5;
    const int lane = tid & 31;
    const int lc   = lane << 2;   // 4 channels per lane

    bool waited = false;
    for (int rep = 0; rep < 2; ++rep) {
        const int i = blockIdx.x + (rep << 8);

        // frame of residue i (uniform per block-half; overlaps DMA on rep 0)
        const float* fr = frames_g + i * 9;
        float xx = fr[0], xy = fr[1], xz = fr[2];
        float yx = fr[3], yy = fr[4], yz = fr[5];
        float zx = fr[6], zy = fr[7], zz = fr[8];

        if (!waited) {
            asm volatile("s_wait_asynccnt 0x0" ::: "memory");
            __syncthreads();
            waited = true;
        }

        const float* ca_s = smem + WPACK_FLOATS;
        float cix = ca_s[i * 3 + 0];
        float ciy = ca_s[i * 3 + 1];
        float ciz = ca_s[i * 3 + 2];

        for (int kk = 0; kk < 16; ++kk) {
            int k = (wave << 4) | kk;
            float dx = ca_s[k * 3 + 0] - cix;   // uniform -> LDS broadcast
            float dy = ca_s[k * 3 + 1] - ciy;
            float dz = ca_s[k * 3 + 2] - ciz;

            // dist uses diff + 1e-6 per component (reference quirk)
            float ex = dx + 1e-6f, ey = dy + 1e-6f, ez = dz + 1e-6f;
            float dist = sqrtf(ex * ex + ey * ey + ez * ez);
            int r0;
            {   // linspace(3,80,128) edges, strict; dist<=3 -> bin 0
                int t = (int)floorf((dist - 3.0f) * (127.0f / 77.0f));
                t = t > 127 ? 127 : t;
                r0 = (dist <= 3.0f) ? 0 : t;
            }
            // projections onto residue i's frame (raw diff, no epsilon)
            float vx = dx * xx + dy * xy + dz * xz;
            float vy = dx * yx + dy * yy + dz * yz;
            float vz = dx * zx + dy * zy + dz * zz;
            int r1 = 128 + dir_bin(vx);
            int r2 = 256 + dir_bin(vy);
            int r3 = 384 + dir_bin(vz);

            // 4-way LDS gather (ds_load_b128/lane, conflict-free across the
            // 64 banks: 32 lanes x 16B = 512B per row) + fused bias
            const v4f a = *(const v4f*)(smem + (r0 << 7) + lc);
            const v4f b = *(const v4f*)(smem + (r1 << 7) + lc);
            const v4f c = *(const v4f*)(smem + (r2 << 7) + lc);
            const v4f d = *(const v4f*)(smem + (r3 << 7) + lc);
            v4f r = (a + b) + (c + d);

            // streaming write-once output: non-temporal b128 store
            v4f* dst = (v4f*)(out + (((size_t)k << 9) + (size_t)i) * 128 + lc);
            __builtin_nontemporal_store(r, dst);
        }
    }
}

extern "C" void kernel_launch(void* const* d_in, const int* in_sizes, int n_in,
                              void* d_out, int out_size, void* d_ws, size_t ws_size,
                              hipStream_t stream) {
    (void)in_sizes; (void)n_in; (void)out_size; (void)ws_size;
    const float* pos = (const float*)d_in[0];  // (1, 512*37, 3)
    const float* W   = (const float*)d_in[1];  // (128, 516)
    const float* b   = (const float*)d_in[2];  // (128,)
    float* out = (float*)d_out;                // (1, 512, 512, 128)

    float* ws       = (float*)d_ws;
    float* wpack    = ws;                      // 65536 floats
    float* ca       = ws + WPACK_FLOATS;       // 1536 floats (contiguous!)
    float* frames_p = ws + STAGE_FLOATS;       // 4608 floats

    pack_w_kernel<<<256, 256, 0, stream>>>(W, b, wpack);
    frames_kernel<<<2, 256, 0, stream>>>(pos, frames_p, ca);
    spe_main_kernel<<<256, 1024, STAGE_FLOATS * sizeof(float), stream>>>(
        ws, frames_p, out);
}